// Renderer_7138235646456
// MI455X (gfx1250) — compile-verified
//
#include <hip/hip_runtime.h>

typedef __attribute__((ext_vector_type(2))) float v2f;
typedef __attribute__((ext_vector_type(8))) float v8f;

#define IMGS   128
#define KTOP   16
#define R2C    0.0004f        // RADIUS^2, RADIUS=0.02
#define INVR2  2500.0f        // 1/R2
#define ZNEARC 1.0f
#define BIGZ   1e30f          // top-K pad
#define BIGD   3e38f          // "invisible" bias -> d2 never < R2
#define FOCAL  1.7320508075688772f  // 1/tan(30deg)

// ---------------------------------------------------------------------------
// Phase 0: project all points. out4[i] = {x_ndc, y_ndc, z_view, x^2+y^2 | BIG}
// ---------------------------------------------------------------------------
__global__ __launch_bounds__(256) void project_kernel(
    const float* __restrict__ pts, const float* __restrict__ eye,
    float4* __restrict__ out4, int npts, int total) {
  int i = blockIdx.x * blockDim.x + threadIdx.x;
  if (i >= total) return;
  int b = i / npts;
  float ex = eye[b * 3 + 0], ey = eye[b * 3 + 1], ez = eye[b * 3 + 2];
  // look_at: z = -eye/|eye|; x = normalize(cross(up,z)); y = cross(z,x)
  float zinv = rsqrtf(ex * ex + ey * ey + ez * ez);
  float zx = -ex * zinv, zy = -ey * zinv, zz = -ez * zinv;
  float xx = zz, xz = -zx;                       // cross((0,1,0), z) = (z.z, 0, -z.x)
  float xinv = rsqrtf(xx * xx + xz * xz);
  xx *= xinv; xz *= xinv;
  float yx = zy * xz;                            // cross(z, x), x.y == 0
  float yy = zz * xx - zx * xz;
  float yz = -zy * xx;
  float dx = pts[i * 3 + 0] - ex, dy = pts[i * 3 + 1] - ey, dz = pts[i * 3 + 2] - ez;
  float cx = dx * xx + dz * xz;                  // x axis has zero y comp
  float cy = dx * yx + dy * yy + dz * yz;
  float cz = dx * zx + dy * zy + dz * zz;
  float zs = (cz > 0.0f) ? cz : 1.0f;
  float xn = cx * FOCAL / zs, yn = cy * FOCAL / zs;
  bool vis = cz > ZNEARC;
  float4 o;
  o.x = vis ? xn : 0.0f;
  o.y = vis ? yn : 0.0f;
  o.z = cz;
  o.w = vis ? (xn * xn + yn * yn) : BIGD;
  out4[i] = o;
}

// Rare-path: scan 8 accumulators of one WMMA, insert hits into sorted top-K.
// Guarded by a single any-hit min-reduce so the common path does one branch.
#define PROC_TILE(CV, BASE)                                                    \
  {                                                                            \
    float mn = fminf(fminf(fminf(CV[0], CV[1]), fminf(CV[2], CV[3])),          \
                     fminf(fminf(CV[4], CV[5]), fminf(CV[6], CV[7])));         \
    if (mn < R2C) {                                                            \
      _Pragma("unroll") for (int r = 0; r < 8; ++r) {                          \
        if (CV[r] < R2C) {                                                     \
          int j = (BASE) + half * 8 + r;                                       \
          float zj = P[j].z;                                                   \
          if (zj < kz[KTOP - 1]) {                                             \
            _Pragma("unroll") for (int t = KTOP - 1; t > 0; --t)               \
                if (kz[t - 1] > zj) { kz[t] = kz[t - 1]; ki[t] = ki[t - 1]; }  \
            _Pragma("unroll") for (int t = KTOP - 1; t >= 0; --t) {            \
              bool pl = (t == 0 || kz[t - 1] <= zj) && (kz[t] > zj);           \
              if (pl) { kz[t] = zj; ki[t] = j; }                               \
            }                                                                  \
          }                                                                    \
        }                                                                      \
      }                                                                        \
    }                                                                          \
  }

// ---------------------------------------------------------------------------
// Phase 1: one wave per 16-pixel tile. d2[point16 x pixel16] via
// v_wmma_f32_16x16x4_f32 (x2 unrolled); per-lane sorted top-K over its
// 8-point rows; lane-pair merge through LDS; alpha composite.
// ---------------------------------------------------------------------------
__global__ __launch_bounds__(256) void render_kernel(
    const float4* __restrict__ pts4, const float* __restrict__ colors,
    float* __restrict__ out, int npts, int tiles_per_batch) {
  __shared__ float s_z[256 * KTOP];
  __shared__ int   s_i[256 * KTOP];

  const int lane = threadIdx.x & 31;
  const int wave = threadIdx.x >> 5;
  const int l16  = lane & 15;
  const int half = lane >> 4;

  const int tile = blockIdx.x * 8 + wave;
  const int b    = tile / tiles_per_batch;
  const int tib  = tile - b * tiles_per_batch;
  const int p    = tib * 16 + l16;               // pixel linear index in batch
  const int row  = p >> 7, col = p & (IMGS - 1);
  const float px = 1.0f - (2.0f * (float)col + 1.0f) / (float)IMGS;
  const float py = 1.0f - (2.0f * (float)row + 1.0f) / (float)IMGS;

  // B operand (4 x 16 pixels): rows [1, px, py, px^2+py^2]
  // lanes 0-15: V0=K0, V1=K1 ; lanes 16-31: V0=K2, V1=K3
  v2f bm;
  bm.x = half ? py : 1.0f;
  bm.y = half ? (px * px + py * py) : px;

  const float4* __restrict__ P = pts4 + (size_t)b * npts;

  float kz[KTOP];
  int   ki[KTOP];
#pragma unroll
  for (int t = 0; t < KTOP; ++t) { kz[t] = BIGZ; ki[t] = 0; }

  for (int cb = 0; cb < npts; cb += 32) {        // npts % 32 == 0
    // speculative (DEV-scope) prefetch: safe past end, silently dropped
    __builtin_prefetch(&P[cb + 64 + l16], 0, 1);
    float4 pt0 = P[cb + l16];
    float4 pt1 = P[cb + 16 + l16];
    // A operand (16 points x 4): rows [x^2+y^2|BIG, -2x, -2y, 1]
    v2f a0, a1;
    a0.x = half ? (-2.0f * pt0.y) : pt0.w;
    a0.y = half ? 1.0f : (-2.0f * pt0.x);
    a1.x = half ? (-2.0f * pt1.y) : pt1.w;
    a1.y = half ? 1.0f : (-2.0f * pt1.x);
    v8f c0 = {0.f, 0.f, 0.f, 0.f, 0.f, 0.f, 0.f, 0.f};
    v8f c1 = {0.f, 0.f, 0.f, 0.f, 0.f, 0.f, 0.f, 0.f};
    c0 = __builtin_amdgcn_wmma_f32_16x16x4_f32(
        false, a0, false, bm, (short)0, c0, false, false);
    c1 = __builtin_amdgcn_wmma_f32_16x16x4_f32(
        false, a1, false, bm, (short)0, c1, false, false);
    // c[r] = d2[point base + 8*half + r][pixel l16]
    PROC_TILE(c0, cb)
    PROC_TILE(c1, cb + 16)
  }

  // spill both half-lists so the merge can index dynamically
  const int base = threadIdx.x * KTOP;
#pragma unroll
  for (int t = 0; t < KTOP; ++t) { s_z[base + t] = kz[t]; s_i[base + t] = ki[t]; }
  __syncthreads();

  if (half == 0) {
    const int baseA = base;
    const int baseB = (threadIdx.x + 16) * KTOP;
    int ia = 0, ib = 0;
    float T = 1.0f, rr = 0.0f, gg = 0.0f, bb = 0.0f;
#pragma unroll
    for (int k = 0; k < KTOP; ++k) {
      float za = s_z[baseA + ia], zb = s_z[baseB + ib];
      bool ta = (za <= zb);
      float zc = ta ? za : zb;
      if (zc >= 1e29f) break;                  // pad -> no more hits
      int id = ta ? s_i[baseA + ia] : s_i[baseB + ib];
      ia += ta ? 1 : 0;
      ib += ta ? 0 : 1;
      float4 pt = P[id];
      float dx = px - pt.x, dy = py - pt.y;
      float d2 = dx * dx + dy * dy;
      float a = 1.0f - d2 * INVR2;
      a = fminf(fmaxf(a, 0.0f), 1.0f);
      float w = a * T;
      const float* cp = colors + ((size_t)b * npts + id) * 3;
      rr += w * cp[0]; gg += w * cp[1]; bb += w * cp[2];
      T *= (1.0f - a);
    }
    float* o = out + ((size_t)b * (IMGS * IMGS) + p) * 3;
    o[0] = rr; o[1] = gg; o[2] = bb;
  }
}

// ---------------------------------------------------------------------------
extern "C" void kernel_launch(void* const* d_in, const int* in_sizes, int n_in,
                              void* d_out, int out_size, void* d_ws, size_t ws_size,
                              hipStream_t stream) {
  const float* points = (const float*)d_in[0];   // [B,N,3] f32
  const float* eye    = (const float*)d_in[1];   // [B,3]   f32
  const float* colors = (const float*)d_in[2];   // [B,N,3] f32
  float* out = (float*)d_out;                    // [B,128,128,3] f32
  float4* pts4 = (float4*)d_ws;                  // B*N * 16 bytes

  const int B = in_sizes[1] / 3;                 // 2
  const int total = in_sizes[0] / 3;             // B*N = 4096
  const int N = total / B;                       // 2048

  int pblocks = (total + 255) / 256;
  project_kernel<<<pblocks, 256, 0, stream>>>(points, eye, pts4, N, total);

  const int tiles_per_batch = (IMGS * IMGS) / 16;      // 1024
  const int tiles = B * tiles_per_batch;               // 2048
  const int rblocks = tiles / 8;                       // 256 blocks x 8 waves
  render_kernel<<<rblocks, 256, 0, stream>>>(pts4, colors, out, N, tiles_per_batch);
}